// ResidualVQ_47734266528358
// MI455X (gfx1250) — compile-verified
//
#include <hip/hip_runtime.h>

typedef float v2f __attribute__((ext_vector_type(2)));
typedef float v8f __attribute__((ext_vector_type(8)));

#define B_N 65536
#define D_N 512
#define R_N 6
#define K_N 1024
#define BM 64
#define NT_MAIN 128

static constexpr size_t OFF_CODES = 0;
static constexpr size_t OFF_ZQ    = (size_t)B_N * R_N;                 // 393216
static constexpr size_t OFF_RES   = OFF_ZQ + (size_t)B_N * D_N;        // 33947648
static constexpr size_t OFF_LOSS  = OFF_RES + (size_t)B_N * D_N;       // 67502080
static constexpr size_t OFF_USAGE = OFF_LOSS + 1;                      // 67502081

// ---------------- init: zero usage counts (must be re-zeroed every call) ---
__global__ __launch_bounds__(256) void rvq_init(unsigned* __restrict__ counts) {
  int i = blockIdx.x * 256 + threadIdx.x;
  if (i < R_N * K_N) counts[i] = 0u;
}

// ---------------- precompute ||e_k||^2 per codebook row ---------------------
__global__ __launch_bounds__(256) void rvq_ce(const float* __restrict__ embed,
                                              float* __restrict__ ce) {
  const int wave = threadIdx.x >> 5, lane = threadIdx.x & 31;
  const int row = blockIdx.x * 8 + wave;
  if (row >= R_N * K_N) return;
  const float* e = embed + (size_t)row * D_N;
  float s = 0.f;
  for (int d = lane; d < D_N; d += 32) { float v = e[d]; s += v * v; }
  #pragma unroll
  for (int off = 16; off; off >>= 1) s += __shfl_xor(s, off, 32);
  if (lane == 0) ce[row] = s;
}

// ---------------- fused 6-stage residual VQ --------------------------------
// Block: 128 threads (4 wave32), 64 rows of z. Residual lives in LDS in WMMA
// A-fragment layout for the whole kernel. Codebook chunks (16 codes x 128
// depth cols, B-fragment layout) are double-buffered in LDS and filled with
// GLOBAL_LOAD_ASYNC_TO_LDS_B64 so the DMA overlaps the WMMA inner loop.
__global__ __launch_bounds__(NT_MAIN) void rvq_main(
    const float* __restrict__ z, const float* __restrict__ embed,
    const float* __restrict__ ce, float* __restrict__ out,
    unsigned* __restrict__ counts, float* __restrict__ partials) {
  // fragment layout: elem((mt,d4,lane,v)) = M[mt*16 + (lane&15)][d4*4 + (lane>>4)*2 + v]
  __shared__ float res[BM * D_N];   // 128 KB, A fragments, mt == wave
  __shared__ float cb[2][2048];     // 2 x 8 KB, B fragments, quarter-depth chunk
  __shared__ int   codes_sh[BM];
  __shared__ float wred[4];

  const int t     = threadIdx.x;
  const int wave  = t >> 5;
  const int lane  = t & 31;
  const int lh    = lane >> 4;       // lane half
  const int lm    = lane & 15;
  const int b0    = blockIdx.x * BM;
  const int myrow = wave * 16 + lm;  // row this lane touches in frag layout
  const int cbase = lh * 2;          // depth sub-offset within the 4-deep frag

  // Async gather of one codebook chunk into B-fragment layout in LDS.
  // chunk = kt*4 + h : k-tile kt (16 codes), depth quarter h (32 d4-steps).
  // Thread t issues 8 x b64: slots s = t + 128*i -> d4c = (t>>5)+4*i, lane slot = t&31.
  auto issue_chunk = [&](const float* E, int chunk, float* buf) {
    const int kt = chunk >> 2, h = chunk & 3;
    const float* g = E + (size_t)(kt * 16 + lm) * D_N + (h * 32 + (t >> 5)) * 4 + cbase;
    unsigned l = (unsigned)(size_t)(void*)(buf + (((t >> 5) * 32 + lane) * 2));
    #pragma unroll
    for (int i = 0; i < 8; ++i) {
      asm volatile("global_load_async_to_lds_b64 %0, %1, off"
                   :: "v"(l), "v"(g) : "memory");
      g += 16;    // 4 d4-steps * 4 cols   (64 B in global)
      l += 1024;  // 4 d4-steps * 64 floats (1 KB in LDS)
    }
  };

  // ---- load z tile into LDS, A-fragment layout (conflict-free b64 stores) --
  {
    const float* zr = z + (size_t)(b0 + myrow) * D_N + cbase;
    float* rp = res + wave * 8192 + lane * 2;   // ((wave*128+d4)*32+lane)*2
    #pragma unroll 4
    for (int d4 = 0; d4 < 128; ++d4)
      *(v2f*)(rp + d4 * 64) = *(const v2f*)(zr + d4 * 4);
  }
  __syncthreads();

  for (int r = 0; r < R_N; ++r) {
    const float* E = embed + (size_t)r * K_N * D_N;

    float minv[8];
    int   mini[8];
    #pragma unroll
    for (int j = 0; j < 8; ++j) { minv[j] = 3.0e38f; mini[j] = K_N; }

    v8f acc = {0.f, 0.f, 0.f, 0.f, 0.f, 0.f, 0.f, 0.f};

    issue_chunk(E, 0, cb[0]);                       // prologue prefetch
    for (int chunk = 0; chunk < 256; ++chunk) {     // 64 k-tiles x 4 quarters
      const int kt = chunk >> 2, h = chunk & 3;
      const float* curb = cb[chunk & 1];

      asm volatile("s_wait_asynccnt 0x0" ::: "memory");  // my slice arrived
      __syncthreads();                                   // all slices arrived;
                                                         // other buffer free
      if (chunk + 1 < 256) issue_chunk(E, chunk + 1, cb[(chunk + 1) & 1]);

      if (h == 0) acc = (v8f){0.f, 0.f, 0.f, 0.f, 0.f, 0.f, 0.f, 0.f};

      const float* ra = res + (size_t)(wave * 128 + h * 32) * 64 + lane * 2;
      #pragma unroll 8
      for (int d4 = 0; d4 < 32; ++d4) {
        v2f a = *(const v2f*)(ra + d4 * 64);
        v2f b = *(const v2f*)(curb + d4 * 64 + lane * 2);
        // D = A(16x4,f32) * B(4x16,f32) + C
        acc = __builtin_amdgcn_wmma_f32_16x16x4_f32(
            false, a, false, b, (short)0, acc, false, false);
      }

      if (h == 3) {
        // dist = ||res||^2 + ||e||^2 - 2*dot ; row-constant ||res||^2 dropped
        const int   k   = kt * 16 + lm;      // column this lane holds
        const float cek = ce[r * K_N + k];
        #pragma unroll
        for (int j = 0; j < 8; ++j) {
          float val = cek - 2.0f * acc[j];
          if (val < minv[j] || (val == minv[j] && k < mini[j])) {
            minv[j] = val; mini[j] = k;
          }
        }
      }
    }

    // argmin across the 16 lanes of each half (C/D layout: row = j + 8*lh)
    #pragma unroll
    for (int off = 1; off < 16; off <<= 1) {
      #pragma unroll
      for (int j = 0; j < 8; ++j) {
        float ov = __shfl_xor(minv[j], off, 32);
        int   oi = __shfl_xor(mini[j], off, 32);
        if (ov < minv[j] || (ov == minv[j] && oi < mini[j])) {
          minv[j] = ov; mini[j] = oi;
        }
      }
    }

    if (lm == 0) {
      #pragma unroll
      for (int j = 0; j < 8; ++j) {
        const int row  = wave * 16 + lh * 8 + j;
        const int code = mini[j];
        codes_sh[row] = code;
        out[OFF_CODES + (size_t)(b0 + row) * R_N + r] = (float)code;
        atomicAdd(&counts[r * K_N + code], 1u);
      }
    }
    __syncthreads();

    // residual -= selected codebook row (frag layout, conflict-free)
    {
      const int    code = codes_sh[myrow];
      const float* er   = E + (size_t)code * D_N + cbase;
      float*       rp   = res + wave * 8192 + lane * 2;
      #pragma unroll 4
      for (int d4 = 0; d4 < 128; ++d4) {
        v2f rv = *(const v2f*)(rp + d4 * 64);
        rv -= *(const v2f*)(er + d4 * 4);
        *(v2f*)(rp + d4 * 64) = rv;
      }
    }
    __syncthreads();
  }

  // ---- epilogue: z_q = z - residual ; loss partial = sum(residual^2) ------
  float lsum = 0.f;
  {
    const size_t gbase = (size_t)(b0 + myrow) * D_N + cbase;
    const float* rp    = res + wave * 8192 + lane * 2;
    #pragma unroll 4
    for (int d4 = 0; d4 < 128; ++d4) {
      v2f rv = *(const v2f*)(rp + d4 * 64);
      v2f zv = *(const v2f*)(z + gbase + d4 * 4);
      v2f qv = zv - rv;
      *(v2f*)(out + OFF_ZQ + gbase + d4 * 4)  = qv;
      *(v2f*)(out + OFF_RES + gbase + d4 * 4) = rv;
      lsum += rv.x * rv.x + rv.y * rv.y;
    }
  }
  #pragma unroll
  for (int off = 16; off; off >>= 1) lsum += __shfl_xor(lsum, off, 32);
  if (lane == 0) wred[wave] = lsum;
  __syncthreads();
  if (t == 0) partials[blockIdx.x] = wred[0] + wred[1] + wred[2] + wred[3];
}

// ---------------- finalize: usage + deterministic loss reduction ------------
__global__ __launch_bounds__(256) void rvq_final(const unsigned* __restrict__ counts,
                                                 const float* __restrict__ partials,
                                                 float* __restrict__ out) {
  const int t = threadIdx.x;
  for (int i = t; i < R_N * K_N; i += 256)
    out[OFF_USAGE + i] = (float)counts[i] * (1.0f / (float)B_N);

  float s = 0.f;
  for (int i = t; i < (B_N / BM); i += 256) s += partials[i];
  #pragma unroll
  for (int off = 16; off; off >>= 1) s += __shfl_xor(s, off, 32);
  __shared__ float sm[8];
  if ((t & 31) == 0) sm[t >> 5] = s;
  __syncthreads();
  if (t == 0) {
    float tot = 0.f;
    #pragma unroll
    for (int w = 0; w < 8; ++w) tot += sm[w];
    out[OFF_LOSS] = tot * (1.0f / ((float)B_N * (float)D_N));
  }
}

extern "C" void kernel_launch(void* const* d_in, const int* in_sizes, int n_in,
                              void* d_out, int out_size, void* d_ws, size_t ws_size,
                              hipStream_t stream) {
  (void)in_sizes; (void)n_in; (void)out_size; (void)ws_size;
  const float* z     = (const float*)d_in[0];
  const float* embed = (const float*)d_in[1];
  float*       out   = (float*)d_out;

  float*    ce       = (float*)d_ws;                                          // 6144 f
  unsigned* counts   = (unsigned*)((char*)d_ws + (size_t)R_N * K_N * 4);      // 6144 u32
  float*    partials = (float*)((char*)d_ws + (size_t)R_N * K_N * 8);         // 1024 f

  rvq_init<<<(R_N * K_N + 255) / 256, 256, 0, stream>>>(counts);
  rvq_ce<<<(R_N * K_N) / 8, 256, 0, stream>>>(embed, ce);
  rvq_main<<<B_N / BM, NT_MAIN, 0, stream>>>(z, embed, ce, out, counts, partials);
  rvq_final<<<1, 256, 0, stream>>>(counts, partials, out);
}